// HWDownsampling_12481174962520
// MI455X (gfx1250) — compile-verified
//
#include <hip/hip_runtime.h>
#include <hip/hip_bf16.h>

typedef float v8f __attribute__((ext_vector_type(8)));
typedef float v2f __attribute__((ext_vector_type(2)));

#define BN_EPS 1e-5f

// ---------------------------------------------------------------------------
// Kernel 1: fold Haar butterfly + BN scale into the WMMA-A operand, packed in
// the exact 16x4 f32 A lane layout so the GEMM's A loads are fully coalesced.
//   k = 4c + q, q: 0=(0,0) 1=(0,1) 2=(1,0) 3=(1,1) pixel of the 2x2 quad
//   A lane layout (16x4 f32): lane = (kpairSel)*16 + M, 2 consecutive K per lane
//   Apacked[((mt*64 + s)*32 + lane)*2 + j]  with s = k>>2 (K-step), j = k&1
// ---------------------------------------------------------------------------
__global__ void pack_weights_kernel(const float* __restrict__ W,
                                    const float* __restrict__ gamma,
                                    const float* __restrict__ var,
                                    float* __restrict__ Apacked) {
  int tid = blockIdx.x * blockDim.x + threadIdx.x; // 128*256 = 32768 threads
  int o = tid >> 8;        // out channel 0..127
  int k = tid & 255;       // folded K index 0..255
  int c = k >> 2;          // input channel
  int q = k & 3;           // quad pixel
  const float* Wr = W + o * 256;
  float wll = Wr[c];
  float wlh = Wr[64 + c];
  float whl = Wr[128 + c];
  float whh = Wr[192 + c];
  float s1 = (q < 2) ? 1.f : -1.f;             // lh sign
  float s2 = ((q & 1) == 0) ? 1.f : -1.f;      // hl sign
  float s3 = (q == 0 || q == 3) ? 1.f : -1.f;  // hh sign
  float scale = gamma[o] * rsqrtf(var[o] + BN_EPS);
  float val = 0.5f * scale * (wll + s1 * wlh + s2 * whl + s3 * whh);

  int mt   = o >> 4;                 // M tile (wave) 0..7
  int mrow = o & 15;                 // M within tile
  int lane = (((k >> 1) & 1) << 4) + mrow;
  int j    = k & 1;
  Apacked[(((mt * 64 + c) * 32) + lane) * 2 + j] = val;
}

// ---------------------------------------------------------------------------
// Kernel 2: fused Haar-DWT + 1x1 conv + BN + ReLU as a K=256 GEMM.
// Block = 256 threads (8 wave32). One block -> all 128 out channels x 32
// output columns at one (batch, h). Each wave owns one 16-row M tile and TWO
// 16-col N tiles: every A fragment feeds two WMMAs (halves A/L2 traffic).
// A loads use a fixed per-lane VGPR address + immediate byte offsets (s*256),
// so the inner loop is pure {global_load_b64, ds_load, 2x v_wmma}.
// ---------------------------------------------------------------------------
__global__ __launch_bounds__(256)
void haar_gemm_kernel(const float* __restrict__ x,
                      const float* __restrict__ Apacked,
                      const float* __restrict__ bvec,
                      const float* __restrict__ gamma,
                      const float* __restrict__ beta,
                      const float* __restrict__ mean,
                      const float* __restrict__ var,
                      float* __restrict__ out) {
  __shared__ float sB[32 * 260];   // B operand transposed: sB[n*260 + k], pad 4
  __shared__ float sBias[128];     // folded bias: scale*b + beta - mean*scale

  const int bb = blockIdx.z;       // batch 0..15
  const int h  = blockIdx.y;       // output row 0..127
  const int w0 = blockIdx.x << 5;  // output col tile base (32 cols)

  // ---- stage x: 64 ch x 2 rows x 64 cols = 8192 floats = 32 KB ----
  const int colBase = w0 * 2;
  #pragma unroll
  for (int i = 0; i < 8; ++i) {
    int t   = threadIdx.x + (i << 8);  // 0..2047 float4 slots
    int c   = t >> 5;                  // input channel
    int rem = t & 31;
    int r   = rem >> 4;                // quad row 0/1
    int j   = (rem & 15) << 2;         // col offset within 64 (float4 aligned)
    const float4 v = *(const float4*)(
        x + (((size_t)(bb * 64 + c) * 256 + (2 * h + r)) * 256 + colBase + j));
    int k = (c << 2) + (r << 1);       // K index for (c, q=2r+0)
    int n = j >> 1;                    // output column within tile (0..30)
    float* p0 = &sB[n * 260 + k];
    p0[0] = v.x; p0[1] = v.y;          // (n,   q=2r), (n,   q=2r+1)
    float* p1 = &sB[(n + 1) * 260 + k];
    p1[0] = v.z; p1[1] = v.w;          // (n+1, q=2r), (n+1, q=2r+1)
  }
  if (threadIdx.x < 128) {
    int o = threadIdx.x;
    float sc = gamma[o] * rsqrtf(var[o] + BN_EPS);
    sBias[o] = sc * bvec[o] + beta[o] - mean[o] * sc;
  }
  __syncthreads();

  // ---- GEMM: wave = one 16-row M tile x two 16-col N tiles, K = 64 x k4 ----
  const int lane = threadIdx.x & 31;
  const int wv   = threadIdx.x >> 5;     // wave id == M tile 0..7
  const int nl   = lane & 15;            // column within N sub-tile
  const int kp   = (lane >> 4) << 1;     // K pair select (0 or 2)
  // fixed per-lane A base; inner-loop loads use immediate offsets s*256 B
  const float* aPtr = Apacked + ((size_t)(wv * 64) * 32 + lane) * 2;
  __builtin_prefetch(aPtr, 0, 1);        // global_prefetch_b8 on A stream
  const float* b0Ptr = &sB[nl * 260 + kp];
  const float* b1Ptr = &sB[(nl + 16) * 260 + kp];

  v8f acc0 = {0.f, 0.f, 0.f, 0.f, 0.f, 0.f, 0.f, 0.f};
  v8f acc1 = {0.f, 0.f, 0.f, 0.f, 0.f, 0.f, 0.f, 0.f};
  #pragma unroll 8
  for (int s = 0; s < 64; ++s) {
    v2f a  = *(const v2f*)(aPtr + s * 64);      // global_load_b64, imm offset
    v2f f0 = *(const v2f*)(b0Ptr + (s << 2));   // ds_load_b64, conflict-free
    v2f f1 = *(const v2f*)(b1Ptr + (s << 2));
    acc0 = __builtin_amdgcn_wmma_f32_16x16x4_f32(
        false, a, false, f0, (short)0, acc0, false, false);
    acc1 = __builtin_amdgcn_wmma_f32_16x16x4_f32(
        false, a, false, f1, (short)0, acc1, false, false);
  }

  // ---- epilogue: bias + ReLU, coalesced stores ----
  // C/D 16x16 f32 layout: VGPR v holds M = v + 8*(lane>=16), N = lane&15
  const int oBase = (wv << 4) + ((lane >> 4) << 3);
  float* outp = out + (((size_t)(bb * 128 + oBase) * 128 + h) * 128 + w0 + nl);
  #pragma unroll
  for (int v = 0; v < 8; ++v) {
    float bias = sBias[oBase + v];
    float r0 = fmaxf(acc0[v] + bias, 0.f);
    float r1 = fmaxf(acc1[v] + bias, 0.f);
    outp[(size_t)v * 128 * 128]      = r0;
    outp[(size_t)v * 128 * 128 + 16] = r1;
  }
}

extern "C" void kernel_launch(void* const* d_in, const int* in_sizes, int n_in,
                              void* d_out, int out_size, void* d_ws, size_t ws_size,
                              hipStream_t stream) {
  const float* x     = (const float*)d_in[0];  // (16, 64, 256, 256)
  const float* W     = (const float*)d_in[1];  // (128, 256)
  const float* b     = (const float*)d_in[2];  // (128,)
  const float* gamma = (const float*)d_in[3];
  const float* beta  = (const float*)d_in[4];
  const float* mean  = (const float*)d_in[5];
  const float* var   = (const float*)d_in[6];
  float* out = (float*)d_out;                  // (16, 128, 128, 128)

  float* Apacked = (float*)d_ws;               // 128*256 floats = 128 KB

  pack_weights_kernel<<<128, 256, 0, stream>>>(W, gamma, var, Apacked);

  dim3 grid(4, 128, 16);                       // (32-col w tiles, h, batch)
  haar_gemm_kernel<<<grid, 256, 0, stream>>>(x, Apacked, b, gamma, beta, mean,
                                             var, out);
}